// LGCN_5497558139658
// MI455X (gfx1250) — compile-verified
//
#include <hip/hip_runtime.h>

#define Bz 4
#define Tz 512
#define Cz 128
#define NEG_INF (-1e22f)

typedef __attribute__((ext_vector_type(2))) float v2f;
typedef __attribute__((ext_vector_type(8))) float v8f;

// ---------------------------------------------------------------------------
// Kernel 1: q = x@WQ, k = x@WK.  x flattened to (B*T, C).
// One wave (32 threads) computes one 16x16 tile of both q and k using
// V_WMMA_F32_16X16X4_F32, sharing the A-matrix (x) loads.
//
// fp32 WMMA layouts (ISA 7.12.2):
//   A 16x4 : lane L, vgpr v -> A[M = L&15, K = v + 2*(L>>4)]
//   B 4x16 : lane L, vgpr v -> B[K = v + 2*(L>>4), N = L&15]
//   C 16x16: lane L, vgpr v -> C[M = v + 8*(L>>4), N = L&15]
// ---------------------------------------------------------------------------
__global__ __launch_bounds__(32)
void lgcn_qk_gemm(const float* __restrict__ x, const float* __restrict__ WQ,
                  const float* __restrict__ WK, float* __restrict__ q,
                  float* __restrict__ k) {
  const int lane = threadIdx.x;            // 0..31
  const int col0 = blockIdx.x * 16;        // output-column tile
  const int row0 = blockIdx.y * 16;        // row tile over B*T
  const int m    = lane & 15;
  const int kh   = (lane >> 4) * 2;        // 0 or 2

  const float* xr = x + (size_t)(row0 + m) * Cz;
  v8f accq = {};
  v8f acck = {};
  for (int k0 = 0; k0 < Cz; k0 += 4) {
    v2f a;                                  // A[v] = x[row][k0 + kh + v]
    a.x = xr[k0 + kh + 0];
    a.y = xr[k0 + kh + 1];
    const int kb = (k0 + kh) * Cz + col0 + m;
    v2f bq, bk;                             // B[v] = W[(k0 + kh + v)*C + col]
    bq.x = WQ[kb];      bq.y = WQ[kb + Cz];
    bk.x = WK[kb];      bk.y = WK[kb + Cz];
    accq = __builtin_amdgcn_wmma_f32_16x16x4_f32(false, a, false, bq,
                                                 (short)0, accq, false, false);
    acck = __builtin_amdgcn_wmma_f32_16x16x4_f32(false, a, false, bk,
                                                 (short)0, acck, false, false);
  }
#pragma unroll
  for (int v = 0; v < 8; ++v) {
    const int row = row0 + v + 8 * (lane >> 4);
    q[(size_t)row * Cz + col0 + m] = accq[v];
    k[(size_t)row * Cz + col0 + m] = acck[v];
  }
}

// ---------------------------------------------------------------------------
// Kernel 2: fused score -> mask -> softmax -> att@x for one (b, 16-row tile).
// 256 threads = 8 full wave32s (EXEC all-1s, as WMMA requires).
// q chunks are double-buffered in LDS via GLOBAL_LOAD_ASYNC_TO_LDS_B128
// (ASYNCcnt-tracked), so chunk jc+1's copy overlaps chunk jc's 128-deep
// FMA reduction.
// ---------------------------------------------------------------------------
#define KS 132                    // 16x128 tile, padded row stride (banks)
#define SS 516                    // 16x512 score tile, padded row stride

__global__ __launch_bounds__(256)
void lgcn_attn(const float* __restrict__ x, const float* __restrict__ adj,
               const float* __restrict__ p, const float* __restrict__ q,
               const float* __restrict__ k, float* __restrict__ out) {
  __shared__ float k_s[16 * KS];
  __shared__ float q_s[2][16 * KS];        // double-buffered q chunks
  __shared__ float p_s[Cz];
  __shared__ float sc[16 * SS];

  const int t    = threadIdx.x;
  const int lane = t & 31;
  const int wid  = t >> 5;                 // 0..7
  const int b    = blockIdx.y;             // batch
  const int i0   = blockIdx.x * 16;        // row tile

  // Issue one chunk's async global->LDS copy: 2048 floats as 2 B128 ops/thread.
  auto issue_q_chunk = [&](int jc, int buf) {
#pragma unroll
    for (int u = 0; u < 2; ++u) {
      const int c   = t + u * 256;         // 0..511 16B-chunk index
      const int row = c >> 5;              // 0..15
      const int c4  = (c & 31) * 4;        // float offset within row, 16B align
      const uint64_t gsrc =
          (uint64_t)(uintptr_t)(q + ((size_t)b * Tz + jc * 16 + row) * Cz + c4);
      const uint32_t ldst =
          (uint32_t)(uintptr_t)(&q_s[buf][row * KS + c4]);
      asm volatile("global_load_async_to_lds_b128 %0, %1, off"
                   :: "v"(ldst), "v"(gsrc)
                   : "memory");
    }
  };

  // ---- stage k tile (16 x 128) and p into LDS; prefetch q chunk 0 ---------
  issue_q_chunk(0, 0);
  for (int idx = t; idx < 16 * Cz; idx += 256) {
    const int i = idx >> 7, d = idx & 127;
    k_s[i * KS + d] = k[((size_t)b * Tz + i0 + i) * Cz + d];
  }
  if (t < Cz) p_s[t] = p[t];

  // ---- score phase: sc[i][j] = p . relu(q[j] + k[i]) + mask ---------------
  const int si = t & 15;                   // local i
  const int sj = t >> 4;                   // local j within chunk
  for (int jc = 0; jc < Tz / 16; ++jc) {
    const int buf = jc & 1;
    if (jc + 1 < Tz / 16) {
      issue_q_chunk(jc + 1, buf ^ 1);      // overlap next copy with compute
      asm volatile("s_wait_asynccnt 0x2" ::: "memory");  // chunk jc resident
    } else {
      asm volatile("s_wait_asynccnt 0x0" ::: "memory");
    }
    __syncthreads();

    const float* kr = &k_s[si * KS];
    const float* qr = &q_s[buf][sj * KS];
    float s = 0.f;
#pragma unroll 8
    for (int d = 0; d < Cz; ++d) {
      const float h = qr[d] + kr[d];
      s = fmaf(p_s[d], fmaxf(h, 0.f), s);
    }
    const int j = jc * 16 + sj;
    const float a = adj[((size_t)b * Tz + i0 + si) * Tz + j];
    sc[si * SS + j] = s + (a > 0.f ? 0.f : NEG_INF);
    __syncthreads();
  }

  // ---- masked softmax over j: wave `wid` handles rows 2*wid, 2*wid+1 ------
  for (int r = 0; r < 2; ++r) {
    const int i = wid * 2 + r;
    float vals[Tz / 32];
    float m = -3.0e38f;
#pragma unroll
    for (int u = 0; u < Tz / 32; ++u) {
      vals[u] = sc[i * SS + lane + 32 * u];
      m = fmaxf(m, vals[u]);
    }
    for (int off = 16; off >= 1; off >>= 1)
      m = fmaxf(m, __shfl_xor(m, off, 32));
    float sum = 0.f;
#pragma unroll
    for (int u = 0; u < Tz / 32; ++u) {
      vals[u] = __expf(vals[u] - m);
      sum += vals[u];
    }
    for (int off = 16; off >= 1; off >>= 1)
      sum += __shfl_xor(sum, off, 32);
    const float inv = 1.0f / sum;
#pragma unroll
    for (int u = 0; u < Tz / 32; ++u)
      sc[i * SS + lane + 32 * u] = vals[u] * inv;
  }
  __syncthreads();

  // ---- out tile = att(16x512) @ x[b](512x128); wave owns 16 columns -------
  const int col0 = wid * 16;
  const int m16  = lane & 15;
  const int kh   = (lane >> 4) * 2;
  const float* xb = x + (size_t)b * Tz * Cz;
  v8f acc = {};
  for (int k0 = 0; k0 < Tz; k0 += 4) {
    v2f a;                                  // A[v] = att[m16][k0 + kh + v]
    a.x = sc[m16 * SS + k0 + kh + 0];
    a.y = sc[m16 * SS + k0 + kh + 1];
    const size_t kb = (size_t)(k0 + kh) * Cz + col0 + m16;
    v2f bb;                                 // B[v] = x[b][k0 + kh + v][col]
    bb.x = xb[kb];
    bb.y = xb[kb + Cz];
    acc = __builtin_amdgcn_wmma_f32_16x16x4_f32(false, a, false, bb,
                                                (short)0, acc, false, false);
  }
#pragma unroll
  for (int v = 0; v < 8; ++v) {
    const int row = i0 + v + 8 * (lane >> 4);
    out[((size_t)b * Tz + row) * Cz + col0 + m16] = acc[v];
  }
}

// ---------------------------------------------------------------------------
extern "C" void kernel_launch(void* const* d_in, const int* in_sizes, int n_in,
                              void* d_out, int out_size, void* d_ws,
                              size_t ws_size, hipStream_t stream) {
  const float* x   = (const float*)d_in[0];  // (B,T,C)
  const float* adj = (const float*)d_in[1];  // (B,T,T)
  const float* WQ  = (const float*)d_in[2];  // (C,C)
  const float* WK  = (const float*)d_in[3];  // (C,C)
  const float* p   = (const float*)d_in[4];  // (C,)
  float* out = (float*)d_out;                // (B,T,C)

  float* qbuf = (float*)d_ws;                       // B*T*C floats (1 MB)
  float* kbuf = qbuf + (size_t)Bz * Tz * Cz;        // B*T*C floats (1 MB)

  dim3 g1(Cz / 16, (Bz * Tz) / 16);                 // (8, 128) tiles
  lgcn_qk_gemm<<<g1, 32, 0, stream>>>(x, WQ, WK, qbuf, kbuf);

  dim3 g2(Tz / 16, Bz);                             // (32, 4) blocks
  lgcn_attn<<<g2, 256, 0, stream>>>(x, adj, p, qbuf, kbuf, out);
}